// SAGEConv_18932215840939
// MI455X (gfx1250) — compile-verified
//
#include <hip/hip_runtime.h>

typedef float v2f __attribute__((ext_vector_type(2)));
typedef float v8f __attribute__((ext_vector_type(8)));

#define DIM  128   // feature dim D and D_OUT
#define KDIM 256   // concat dim 2*D

// ---------------------------------------------------------------------------
// Zero-init workspace (summed accumulator + degree counts)
// ---------------------------------------------------------------------------
__global__ __launch_bounds__(256) void zero_ws(float* __restrict__ p, int n) {
  int i = blockIdx.x * 256 + threadIdx.x;
  if (i < n) p[i] = 0.0f;
}

// ---------------------------------------------------------------------------
// Edge phase: gather h_src rows (float4 / B128 loads) and scatter-add into the
// per-destination accumulator with f32 global atomics (L2-resident: 51MB of
// accumulator << 192MB L2, so atomics never touch HBM). One 32-thread group
// per edge; lane handles 4 contiguous floats. Lane 0 bumps the degree.
// ---------------------------------------------------------------------------
__global__ __launch_bounds__(256) void scatter_mean_accum(
    const float* __restrict__ hsrc, const int* __restrict__ sidx,
    const int* __restrict__ didx, float* __restrict__ summed,
    float* __restrict__ deg, int nE) {
  int t = blockIdx.x * 256 + threadIdx.x;
  int e = t >> 5;
  if (e >= nE) return;
  int sub = t & 31;
  int s = sidx[e];
  int d = didx[e];
  const float4* srow = (const float4*)(hsrc + (size_t)s * DIM);
  float4 v = srow[sub];                       // global_load_b128
  float* drow = summed + (size_t)d * DIM + sub * 4;
  atomicAdd(drow + 0, v.x);
  atomicAdd(drow + 1, v.y);
  atomicAdd(drow + 2, v.z);
  atomicAdd(drow + 3, v.w);
  if (sub == 0) atomicAdd(deg + d, 1.0f);
}

// ---------------------------------------------------------------------------
// Projection: out[16 nodes x 128] = [h_dst | mean] (16x256) @ W (256x128) + b
// using V_WMMA_F32_16X16X4_F32. 256 threads = 8 waves; block stages the
// 16x256 h_total tile in LDS transposed [k][m] (pitch 17 -> conflict-free
// A-fragment reads, lowered to ds_load_2addr_b32); wave w computes output
// columns [16w, 16w+16). Epilogue uses a block-uniform branch so the common
// full-tile path issues 8 unconditional global_store_b32 (no EXEC churn).
// ---------------------------------------------------------------------------
__global__ __launch_bounds__(256) void sage_gemm_wmma(
    const float* __restrict__ hdst, const float* __restrict__ summed,
    const float* __restrict__ deg, const float* __restrict__ W,
    const float* __restrict__ bias, float* __restrict__ out, int ndst) {
  __shared__ float sh[KDIM][17];   // h_total tile, transposed: sh[k][m]
  __shared__ float shInv[16];      // 1 / max(deg, 1) per row

  const int tid = threadIdx.x;
  const int node_base = blockIdx.x * 16;

  // Prefetch the whole 128KB W matrix toward L2/WGP$ (global_prefetch_b8).
  {
    const char* wb = (const char*)W;
#pragma unroll
    for (int j = 0; j < 4; ++j)
      __builtin_prefetch(wb + tid * 128 + j * 32768, 0, 1);
  }

  if (tid < 16) {
    int node = min(node_base + tid, ndst - 1);
    shInv[tid] = 1.0f / fmaxf(deg[node], 1.0f);
  }
  __syncthreads();

  // Fill LDS: iteration j loads row j, columns coalesced across the block.
#pragma unroll
  for (int j = 0; j < 16; ++j) {
    int row = j;
    int col = tid;                 // 0..255
    int node = min(node_base + row, ndst - 1);
    float val;
    if (col < DIM) {
      val = hdst[(size_t)node * DIM + col];
    } else {
      val = summed[(size_t)node * DIM + (col - DIM)] * shInv[row];
    }
    sh[col][row] = val;            // pitch-17 rows: conflict-free scatter
  }
  __syncthreads();

  const int wave = tid >> 5;       // 0..7 -> output column tile
  const int lane = tid & 31;
  const int m    = lane & 15;          // A-matrix row / lane mapping
  const int koff = (lane >> 4) << 1;   // lanes 0-15 -> K+0/1, 16-31 -> K+2/3
  const int nb   = wave * 16;
  const int n_in = lane & 15;          // B/C/D column within tile

  v8f acc = {};                        // 16x16 f32 accumulator (8 VGPRs)

#pragma unroll 8
  for (int k = 0; k < KDIM; k += 4) {
    v2f a;
    a.x = sh[k + koff + 0][m];
    a.y = sh[k + koff + 1][m];
    v2f b;
    const float* wp = W + (size_t)(k + koff) * DIM + nb + n_in;
    b.x = wp[0];
    b.y = wp[DIM];
    // D = A(16x4) * B(4x16) + C  -> v_wmma_f32_16x16x4_f32
    acc = __builtin_amdgcn_wmma_f32_16x16x4_f32(
        /*neg_a=*/false, a, /*neg_b=*/false, b,
        /*c_mod=*/(short)0, acc, /*reuse_a=*/false, /*reuse_b=*/false);
  }

  const float bv = bias[nb + n_in];
  const int mbase = (lane >> 4) << 3;          // 0 or 8
  float* orow = out + (size_t)(node_base + mbase) * DIM + nb + n_in;

  if (node_base + 16 <= ndst) {
    // Fast path (block-uniform): full tile, unconditional stores.
#pragma unroll
    for (int v = 0; v < 8; ++v)
      orow[(size_t)v * DIM] = acc[v] + bv;     // 8x global_store_b32
  } else {
    // Ragged tail tile (at most one block per launch).
#pragma unroll
    for (int v = 0; v < 8; ++v) {
      int node = node_base + mbase + v;
      if (node < ndst)
        out[(size_t)node * DIM + nb + n_in] = acc[v] + bv;
    }
  }
}

// ---------------------------------------------------------------------------
// Host-side launcher
// ---------------------------------------------------------------------------
extern "C" void kernel_launch(void* const* d_in, const int* in_sizes, int n_in,
                              void* d_out, int out_size, void* d_ws, size_t ws_size,
                              hipStream_t stream) {
  const float* h_src  = (const float*)d_in[0];
  const float* h_dst  = (const float*)d_in[1];
  const int*   s_idx  = (const int*)d_in[2];
  const int*   d_idx  = (const int*)d_in[3];
  const float* W      = (const float*)d_in[4];
  const float* b      = (const float*)d_in[5];
  float*       outp   = (float*)d_out;

  const int ndst = in_sizes[1] / DIM;
  const int nE   = in_sizes[2];

  float* summed = (float*)d_ws;                       // [ndst * 128]
  float* deg    = summed + (size_t)ndst * DIM;        // [ndst]

  const int nzero = ndst * DIM + ndst;
  zero_ws<<<(nzero + 255) / 256, 256, 0, stream>>>(summed, nzero);

  const int sthreads = nE * 32;                       // 32 lanes per edge
  scatter_mean_accum<<<(sthreads + 255) / 256, 256, 0, stream>>>(
      h_src, s_idx, d_idx, summed, deg, nE);

  sage_gemm_wmma<<<(ndst + 15) / 16, 256, 0, stream>>>(
      h_dst, summed, deg, W, b, outp, ndst);

  (void)n_in; (void)out_size; (void)ws_size;
}